// ParallelKmeans_66614942761221
// MI455X (gfx1250) — compile-verified
//
#include <hip/hip_runtime.h>

typedef __attribute__((ext_vector_type(2))) float v2f;
typedef __attribute__((ext_vector_type(8))) float v8f;

#define NC   256
#define DIM  64
#define NSP  16
#define NPTS 32768
#define NITER 5

#define BLK_THREADS 256
#define BLOCKS_PER_SPACE 16
#define PTS_PER_BLOCK (NPTS / BLOCKS_PER_SPACE)   // 2048
#define PTS_PER_PASS  128
#define NPASS (PTS_PER_BLOCK / PTS_PER_PASS)      // 16

#define CPAD 68   // centroid LDS row stride (floats) -> conflict-free B-frag reads
#define XPAD 66   // x-tile LDS row stride
#define SPAD 66   // sums LDS row stride

// ---------------------------------------------------------------- init: c0 = x[:256] transposed, plus c2
__global__ void kmeans_init(const float* __restrict__ x, float* __restrict__ c,
                            float* __restrict__ c2) {
    int t = blockIdx.x * blockDim.x + threadIdx.x;   // 0..4095 = (s,k)
    int s = t >> 8, k = t & 255;
    const float* src = x + ((size_t)k * NSP + s) * DIM;
    float*       dst = c + ((size_t)s * NC + k) * DIM;
    float acc = 0.f;
    for (int d = 0; d < DIM; d += 4) {
        float4 v = *(const float4*)(src + d);
        *(float4*)(dst + d) = v;
        acc += v.x * v.x + v.y * v.y + v.z * v.z + v.w * v.w;
    }
    c2[t] = acc;
}

// ---------------------------------------------------------------- zero scratch accumulators
__global__ void kmeans_zero(float* __restrict__ gsum, float* __restrict__ gcnt) {
    int t = blockIdx.x * blockDim.x + threadIdx.x;
    if (t < NSP * NC * DIM) gsum[t] = 0.f;
    if (t < NSP * NC)       gcnt[t] = 0.f;
}

// ---------------------------------------------------------------- fused assign + accumulate (WMMA f32 16x16x4)
__global__ __launch_bounds__(BLK_THREADS)
void kmeans_assign(const float* __restrict__ x, const float* __restrict__ c,
                   const float* __restrict__ c2, float* __restrict__ gsum,
                   float* __restrict__ gcnt) {
    __shared__ float lds_c[NC * CPAD];            // centroids, row-major, padded
    __shared__ float lds_x[PTS_PER_PASS * XPAD];  // staged points
    __shared__ float lds_s[NC * SPAD];            // per-block centroid sums
    __shared__ float lds_cnt[NC];
    __shared__ float lds_c2[NC];
    __shared__ int   lds_asgn[PTS_PER_PASS];

    const int t    = threadIdx.x;
    const int lane = t & 31;
    const int wave = t >> 5;
    const int s    = blockIdx.x / BLOCKS_PER_SPACE;
    const int blk  = blockIdx.x % BLOCKS_PER_SPACE;
    const int pt0  = blk * PTS_PER_BLOCK;

    // load this space's centroids into LDS (float4, CPAD%4==0 keeps 16B alignment)
    const float4* cg = (const float4*)(c + (size_t)s * NC * DIM);
    #pragma unroll
    for (int i = 0; i < 16; ++i) {
        int e = t + i * BLK_THREADS;              // 0..4095 float4s
        int k = e >> 4, q = e & 15;
        float4 v = cg[(size_t)k * 16 + q];
        *(float4*)&lds_c[k * CPAD + q * 4] = v;
    }
    if (t < NC) { lds_c2[t] = c2[s * NC + t]; lds_cnt[t] = 0.f; }
    for (int i = 0; i < SPAD; ++i) lds_s[t * SPAD + i] = 0.f;   // 256*66 == NC*SPAD
    __syncthreads();

    const int lh = lane >> 4;    // half of wave: selects K{0,1} vs K{2,3}
    const int ll = lane & 15;    // M (A-frag) / N (B-frag) within tile
    const int m0 = wave * 16;    // this wave's 16-point tile

    // prologue: prefetch pass 0's x tile into registers
    float4 pre[8];
    #pragma unroll
    for (int i = 0; i < 8; ++i) {
        int e = t + i * BLK_THREADS;              // 0..2047 float4s
        int p = e >> 4, q = e & 15;
        pre[i] = *(const float4*)(x + (((size_t)(pt0 + p)) * NSP + s) * DIM + q * 4);
    }

    for (int pass = 0; pass < NPASS; ++pass) {
        // commit the staged tile to LDS (float2 stores keep 8B alignment under XPAD=66)
        #pragma unroll
        for (int i = 0; i < 8; ++i) {
            int e = t + i * BLK_THREADS;
            int p = e >> 4, q = e & 15;
            float* d = &lds_x[p * XPAD + q * 4];
            *(float2*)(d + 0) = make_float2(pre[i].x, pre[i].y);
            *(float2*)(d + 2) = make_float2(pre[i].z, pre[i].w);
        }
        __syncthreads();

        // prefetch NEXT pass's tile; these global loads overlap the WMMA phase below
        // and are only waited on at the next pass's LDS commit.
        if (pass + 1 < NPASS) {
            const int p0n = pt0 + (pass + 1) * PTS_PER_PASS;
            #pragma unroll
            for (int i = 0; i < 8; ++i) {
                int e = t + i * BLK_THREADS;
                int p = e >> 4, q = e & 15;
                pre[i] = *(const float4*)(x + (((size_t)(p0n + p)) * NSP + s) * DIM + q * 4);
            }
        }

        // A fragments for all 16 k-steps (16x4 f32 layout: lane<16 -> K0,K1; lane>=16 -> K2,K3)
        v2f afrag[16];
        #pragma unroll
        for (int kk = 0; kk < 16; ++kk)
            afrag[kk] = *(const v2f*)&lds_x[(m0 + ll) * XPAD + kk * 4 + 2 * lh];

        float best[8];
        int   bidx[8];
        #pragma unroll
        for (int v = 0; v < 8; ++v) { best[v] = 3.4e38f; bidx[v] = 0; }

        for (int nt = 0; nt < 16; ++nt) {         // 16 centroid tiles of 16
            // two independent accumulator chains -> 2x WMMA ILP
            v8f acc0 = {};
            v8f acc1 = {};
            #pragma unroll
            for (int kk = 0; kk < 8; ++kk) {
                v2f b0 = *(const v2f*)&lds_c[(nt * 16 + ll) * CPAD + (2 * kk)     * 4 + 2 * lh];
                v2f b1 = *(const v2f*)&lds_c[(nt * 16 + ll) * CPAD + (2 * kk + 1) * 4 + 2 * lh];
                acc0 = __builtin_amdgcn_wmma_f32_16x16x4_f32(
                        false, afrag[2 * kk],     false, b0, (short)0, acc0, false, false);
                acc1 = __builtin_amdgcn_wmma_f32_16x16x4_f32(
                        false, afrag[2 * kk + 1], false, b1, (short)0, acc1, false, false);
            }
            const int   n  = nt * 16 + ll;        // N striped across lanes (C/D layout)
            const float cc = lds_c2[n];
            #pragma unroll
            for (int v = 0; v < 8; ++v) {         // M = v + 8*lh
                float score = cc - 2.0f * (acc0[v] + acc1[v]);  // x^2 constant -> dropped
                if (score < best[v]) { best[v] = score; bidx[v] = n; }
            }
        }

        // argmin across the 16 lanes of each half (N dimension)
        #pragma unroll
        for (int v = 0; v < 8; ++v) {
            #pragma unroll
            for (int msk = 1; msk <= 8; msk <<= 1) {
                float ob = __shfl_xor(best[v], msk, 32);
                int   oi = __shfl_xor(bidx[v], msk, 32);
                if (ob < best[v] || (ob == best[v] && oi < bidx[v])) {
                    best[v] = ob; bidx[v] = oi;
                }
            }
        }
        if (ll == 0) {
            #pragma unroll
            for (int v = 0; v < 8; ++v)
                lds_asgn[m0 + lh * 8 + v] = bidx[v];   // point m = v + 8*lh
        }
        __syncthreads();

        // accumulate this pass's points into per-block LDS sums (2 threads per point)
        {
            const int p = t >> 1;
            const int h = t & 1;
            const int a = lds_asgn[p];
            if (h == 0) atomicAdd(&lds_cnt[a], 1.0f);
            const float* xs = &lds_x[p * XPAD + h * 32];
            float*       ss = &lds_s[a * SPAD + h * 32];
            #pragma unroll
            for (int d = 0; d < 32; ++d) atomicAdd(&ss[d], xs[d]);
        }
        __syncthreads();
    }

    // flush block-local sums/counts to global accumulators
    for (int i = 0; i < 64; ++i) {
        int e = t + i * BLK_THREADS;              // 0..16383
        int k = e >> 6, d = e & 63;
        atomicAdd(&gsum[((size_t)s * NC + k) * DIM + d], lds_s[k * SPAD + d]);
    }
    if (t < NC) atomicAdd(&gcnt[s * NC + t], lds_cnt[t]);
}

// ---------------------------------------------------------------- centroid update + c2 for next iter
__global__ void kmeans_update(const float* __restrict__ gsum, const float* __restrict__ gcnt,
                              float* __restrict__ c, float* __restrict__ c2) {
    int t = blockIdx.x * blockDim.x + threadIdx.x;
    if (t >= NSP * NC) return;
    float cnt = gcnt[t];
    float inv = 1.0f / fmaxf(cnt, 1.0f);
    bool  upd = cnt > 0.f;
    float*       cr = c    + (size_t)t * DIM;
    const float* sr = gsum + (size_t)t * DIM;
    float acc = 0.f;
    for (int d = 0; d < DIM; d += 4) {
        float4 cv = *(const float4*)(cr + d);
        float4 sv = *(const float4*)(sr + d);
        float4 nv;
        nv.x = upd ? sv.x * inv : cv.x;
        nv.y = upd ? sv.y * inv : cv.y;
        nv.z = upd ? sv.z * inv : cv.z;
        nv.w = upd ? sv.w * inv : cv.w;
        *(float4*)(cr + d) = nv;
        acc += nv.x * nv.x + nv.y * nv.y + nv.z * nv.z + nv.w * nv.w;
    }
    c2[t] = acc;
}

// ---------------------------------------------------------------- launch
extern "C" void kernel_launch(void* const* d_in, const int* in_sizes, int n_in,
                              void* d_out, int out_size, void* d_ws, size_t ws_size,
                              hipStream_t stream) {
    const float* x = (const float*)d_in[0];
    float* cbuf = (float*)d_out;                 // centroids live in d_out
    float* gsum = (float*)d_ws;                  // 16*256*64 floats
    float* gcnt = gsum + (size_t)NSP * NC * DIM; // 16*256 floats
    float* c2   = gcnt + (size_t)NSP * NC;       // 16*256 floats

    kmeans_init<<<NSP * NC / 256, 256, 0, stream>>>(x, cbuf, c2);
    for (int it = 0; it < NITER; ++it) {
        kmeans_zero<<<(NSP * NC * DIM) / 256, 256, 0, stream>>>(gsum, gcnt);
        kmeans_assign<<<NSP * BLOCKS_PER_SPACE, BLK_THREADS, 0, stream>>>(x, cbuf, c2, gsum, gcnt);
        kmeans_update<<<NSP * NC / 256, 256, 0, stream>>>(gsum, gcnt, cbuf, c2);
    }
}